// TotalLoss_65343632441929
// MI455X (gfx1250) — compile-verified
//
#include <hip/hip_runtime.h>
#include <cstdint>
#include <cstddef>

// ---------------------------------------------------------------------------
// Problem constants (from reference setup_inputs)
// ---------------------------------------------------------------------------
constexpr int Bn = 2;
constexpr int Dd = 160;
constexpr int Hh = 192;
constexpr int Ww = 160;
constexpr int WIN  = 9;
constexpr int HALO = WIN / 2;          // 4

// LNCC tiling
constexpr int TILE_H = 32;
constexpr int TILE_W = 32;
constexpr int IN_H = TILE_H + 2 * HALO; // 40
constexpr int IN_W = TILE_W + 2 * HALO; // 40
constexpr int D_CHUNK = 16;
constexpr int ND = Dd / D_CHUNK;        // 10
constexpr int NTHREADS = 256;           // 8 waves (wave32)
constexpr int NSTEPS = D_CHUNK + 2 * HALO; // 24 planes per block

constexpr int NCC_BLOCKS = (Ww / TILE_W) * (Hh / TILE_H) * (Bn * ND); // 600
constexpr int GRAD_BLOCKS = 2400;

// Dynamic LDS layout (float offsets). Total 59136 floats = 236,544 bytes.
// CDNA5: up to 320 KB LDS per workgroup (impossible on CDNA4's 64 KB/CU).
constexpr int PLANE = IN_H * IN_W;                              // 1600
constexpr int OFF_IN  = 0;                                      // 2 bufs x (F,M)
constexpr int OFF_WS  = OFF_IN + 4 * PLANE;                     // 6400
constexpr int OFF_PC  = OFF_WS + 5 * IN_H * TILE_W;             // 12800
constexpr int OFF_RED = OFF_PC + WIN * 5 * TILE_H * TILE_W;     // 58880
constexpr int SMEM_FLOATS = OFF_RED + NTHREADS;                 // 59136

// ---------------------------------------------------------------------------
// CDNA5 Tensor Data Mover: async DMA of a 2D tile from global -> LDS
// ---------------------------------------------------------------------------
#if defined(__AMDGCN__) && __has_builtin(__builtin_amdgcn_tensor_load_to_lds)
#define HAVE_TDM 1
#endif

#if defined(HAVE_TDM)
typedef unsigned int tdm_u32x4 __attribute__((ext_vector_type(4)));
typedef int          tdm_i32x4 __attribute__((ext_vector_type(4)));
typedef int          tdm_i32x8 __attribute__((ext_vector_type(8)));

// Build a D# (groups 0/1) for a 2D f32 tile load and issue TENSOR_LOAD_TO_LDS.
// Layout per CDNA5 ISA.
__device__ __forceinline__ void tdm_load_2d_f32(uint32_t ldsByteOff,
                                                const float* gaddr,
                                                int tileW, int tileH,
                                                int tensorW, int tensorH,
                                                int rowStrideElems) {
  const uint64_t ga = (uint64_t)(uintptr_t)gaddr;

  tdm_u32x4 g0;
  g0.x = 1u;                              // count=1 (valid), user mode, no gather
  g0.y = ldsByteOff;                      // lds_addr [63:32]
  g0.z = (uint32_t)ga;                    // global_addr[31:0]
  g0.w = (uint32_t)((ga >> 32) & 0x01FFFFFFu) | 0x80000000u; // addr[56:32] | type=2

  tdm_i32x8 g1;
  g1.s0 = (int)(2u << 16);                // workgroup_mask=0 (not in cluster), data_size=4B
  g1.s1 = (int)(((uint32_t)tensorW & 0xFFFFu) << 16);        // tensor_dim0[15:0] @ bit48
  g1.s2 = (int)((((uint32_t)tensorW >> 16) & 0xFFFFu) |
                (((uint32_t)tensorH & 0xFFFFu) << 16));      // dim0[31:16] | dim1[15:0]
  g1.s3 = (int)((((uint32_t)tensorH >> 16) & 0xFFFFu) |
                (((uint32_t)tileW & 0xFFFFu) << 16));        // dim1[31:16] | tile_dim0
  g1.s4 = (int)((uint32_t)tileH & 0xFFFFu);                  // tile_dim1; tile_dim2=0 (2D)
  g1.s5 = (int)rowStrideElems;                               // tensor_dim0_stride[31:0]
  g1.s6 = 0;                                                 // stride hi, dim1_stride lo
  g1.s7 = 0;

  const tdm_i32x4 zero4 = {0, 0, 0, 0};
#if __has_include(<hip/amd_detail/amd_gfx1250_TDM.h>)
  const tdm_i32x8 zero8 = {0, 0, 0, 0, 0, 0, 0, 0};
  __builtin_amdgcn_tensor_load_to_lds(g0, g1, zero4, zero4, zero8, 0);
#else
  __builtin_amdgcn_tensor_load_to_lds(g0, g1, zero4, zero4, 0);
#endif
}

// Tensor ops complete in-order per wave: waiting TENSORcnt<=N with N newer ops
// outstanding guarantees all older ops have landed in LDS.
template <short N>
__device__ __forceinline__ void tdm_wait() {
#if __has_builtin(__builtin_amdgcn_s_wait_tensorcnt)
  __builtin_amdgcn_s_wait_tensorcnt(N);
#else
  asm volatile("s_wait_tensorcnt %0" :: "i"(N) : "memory");
#endif
}
#endif  // HAVE_TDM

// ---------------------------------------------------------------------------
// Fixed-order block reduction (deterministic across graph replays)
// ---------------------------------------------------------------------------
__device__ __forceinline__ float block_reduce_sum(float v, float* red) {
  const int tid = threadIdx.x;
  red[tid] = v;
  __syncthreads();
#pragma unroll
  for (int s = NTHREADS / 2; s > 0; s >>= 1) {
    if (tid < s) red[tid] += red[tid + s];
    __syncthreads();
  }
  const float r = red[0];
  __syncthreads();
  return r;
}

// ---------------------------------------------------------------------------
// Fused LNCC: marching-planes separable 9^3 box sums + cc, single HBM read of
// each image. TDM double-buffers the next plane behind the current compute.
// grid = (W/32, H/32, B*ND), block = 256, dyn LDS = SMEM_FLOATS*4 bytes.
// ---------------------------------------------------------------------------
__global__ __launch_bounds__(NTHREADS)
void lncc_kernel(const float* __restrict__ If, const float* __restrict__ Im,
                 float* __restrict__ ccPart) {
  extern __shared__ float smem[];
  float* sWs  = smem + OFF_WS;    // W-direction 9-sums, 5 fields x 40x32
  float* sPc  = smem + OFF_PC;    // rolling 9-plane HW-window sums, 9 x 5 x 32x32
  float* sRed = smem + OFF_RED;   // reduction scratch

  const int tid = threadIdx.x;
  const int b   = blockIdx.z / ND;
  const int dc  = blockIdx.z % ND;
  const int d0  = dc * D_CHUNK;
  const int h0  = blockIdx.y * TILE_H;
  const int w0  = blockIdx.x * TILE_W;

  const size_t planeStride = (size_t)Hh * Ww;
  const float* baseF = If + (size_t)b * Dd * planeStride;
  const float* baseM = Im + (size_t)b * Dd * planeStride;

  const bool interiorHW = (h0 >= HALO) && (h0 + TILE_H + HALO <= Hh) &&
                          (w0 >= HALO) && (w0 + TILE_W + HALO <= Ww);

  // Issue the load of plane step `i` into double-buffer `buf`.
  // Returns true if the (async) TDM path was used, false if the synchronous
  // zero-padded fallback wrote the buffer directly. Uniform across the block.
  auto plane_load = [&](int i, int buf) -> bool {
    const int d = d0 - HALO + i;
    const bool dValid = (d >= 0) && (d < Dd);
    float* dstF = smem + OFF_IN + buf * (2 * PLANE);
    float* dstM = dstF + PLANE;
#if defined(HAVE_TDM)
    if (dValid && interiorHW) {
      if (tid == 0) {
        const size_t off = (size_t)d * planeStride +
                           (size_t)(h0 - HALO) * Ww + (size_t)(w0 - HALO);
        tdm_load_2d_f32((uint32_t)(uintptr_t)dstF, baseF + off,
                        IN_W, IN_H, Ww, Hh, Ww);
        tdm_load_2d_f32((uint32_t)(uintptr_t)dstM, baseM + off,
                        IN_W, IN_H, Ww, Hh, Ww);
      }
      return true;
    }
#endif
    // Boundary (or no-TDM) path: zero-padded cooperative load (SAME padding).
    for (int t = tid; t < PLANE; t += NTHREADS) {
      const int hh = t / IN_W;
      const int ww = t - hh * IN_W;
      const int h = h0 - HALO + hh;
      const int w = w0 - HALO + ww;
      float f = 0.0f, m = 0.0f;
      if (dValid && (unsigned)h < (unsigned)Hh && (unsigned)w < (unsigned)Ww) {
        const size_t g = (size_t)d * planeStride + (size_t)h * Ww + w;
        f = baseF[g];
        m = baseM[g];
      }
      dstF[t] = f;
      dstM[t] = m;
    }
    return false;
  };

  const float invN = 1.0f / (float)(WIN * WIN * WIN);  // 1/729
  float ccAcc = 0.0f;

  bool curTdm = plane_load(0, 0);  // prime the pipeline

  for (int i = 0; i < NSTEPS; ++i) {
    const int cur = i & 1;
    const float* sF = smem + OFF_IN + cur * (2 * PLANE);
    const float* sM = sF + PLANE;

    // ---- issue next plane's DMA into the other buffer (overlaps compute) ----
    bool nxtTdm = false;
    if (i + 1 < NSTEPS) nxtTdm = plane_load(i + 1, cur ^ 1);

#if defined(HAVE_TDM)
    if (curTdm) {
      if (tid == 0) {
        if (nxtTdm) tdm_wait<2>();  // 2 newer ops in flight; current plane done
        else        tdm_wait<0>();
      }
    }
#endif
    (void)curTdm;
    __syncthreads();  // current plane resident & visible block-wide

    // ---- stage 2: 9-tap sums along W for all 5 fields (products on the fly) ----
    for (int t = tid; t < IN_H * TILE_W; t += NTHREADS) {
      const int hh = t >> 5;          // row in [0,40)
      const int wo = t & (TILE_W - 1);
      const float* rf = sF + hh * IN_W + wo;
      const float* rm = sM + hh * IN_W + wo;
      float s0 = 0, s1 = 0, s2 = 0, s3 = 0, s4 = 0;
#pragma unroll
      for (int k = 0; k < WIN; ++k) {
        const float f = rf[k];
        const float m = rm[k];
        s0 += f; s1 += m; s2 += f * f; s3 += m * m; s4 += f * m;
      }
      sWs[0 * (IN_H * TILE_W) + t] = s0;
      sWs[1 * (IN_H * TILE_W) + t] = s1;
      sWs[2 * (IN_H * TILE_W) + t] = s2;
      sWs[3 * (IN_H * TILE_W) + t] = s3;
      sWs[4 * (IN_H * TILE_W) + t] = s4;
    }
    __syncthreads();

    // ---- stage 3: 9-tap sums along H -> plane HW-window sums into circ slot ----
    {
      float* pcs = sPc + (i % WIN) * (5 * TILE_H * TILE_W);
      for (int t = tid; t < TILE_H * TILE_W; t += NTHREADS) {
        const int ho = t >> 5;
        const int wo = t & (TILE_W - 1);
#pragma unroll
        for (int f = 0; f < 5; ++f) {
          const float* col = sWs + f * (IN_H * TILE_W) + ho * TILE_W + wo;
          float s = 0;
#pragma unroll
          for (int j = 0; j < WIN; ++j) s += col[j * TILE_W];
          pcs[f * (TILE_H * TILE_W) + t] = s;
        }
      }
    }
    __syncthreads();

    // ---- stage 4: once 9 planes buffered, emit cc for output plane d0+i-8 ----
    if (i >= WIN - 1) {
      for (int t = tid; t < TILE_H * TILE_W; t += NTHREADS) {
        float S0 = 0, S1 = 0, S2 = 0, S3 = 0, S4 = 0;
#pragma unroll
        for (int sl = 0; sl < WIN; ++sl) {
          const float* p = sPc + sl * (5 * TILE_H * TILE_W) + t;
          S0 += p[0 * (TILE_H * TILE_W)];
          S1 += p[1 * (TILE_H * TILE_W)];
          S2 += p[2 * (TILE_H * TILE_W)];
          S3 += p[3 * (TILE_H * TILE_W)];
          S4 += p[4 * (TILE_H * TILE_W)];
        }
        const float fm  = S0 * invN;
        const float mm  = S1 * invN;
        const float vf  = S2 - fm * fm;
        const float vm  = S3 - mm * mm;
        const float cov = S4 - fm * mm;
        ccAcc += (cov * cov) / (vf * vm + 1e-8f);
      }
    }
    // Stage-4 reads of sPc slot (i+1)%9 are protected from iteration i+1's
    // stage-3 overwrite by the two barriers that precede it.
    curTdm = nxtTdm;
  }

  const float blockSum = block_reduce_sum(ccAcc, sRed);
  if (tid == 0) {
    const int bid = (blockIdx.z * gridDim.y + blockIdx.y) * gridDim.x + blockIdx.x;
    ccPart[bid] = blockSum;
  }
}

// ---------------------------------------------------------------------------
// Flow gradient loss, float4-vectorized along W (W % 4 == 0, 16B aligned).
// Neighbor re-reads hit L2 (flow = 118 MB < 192 MB global L2).
// ---------------------------------------------------------------------------
__global__ __launch_bounds__(NTHREADS)
void grad_kernel(const float* __restrict__ flow,
                 float* __restrict__ dyP, float* __restrict__ dxP,
                 float* __restrict__ dzP) {
  __shared__ float red[NTHREADS];
  constexpr int NW4 = Ww / 4;                       // 40
  constexpr int N4  = Bn * 3 * Dd * Hh * NW4;       // 7,372,800
  float sdy = 0, sdx = 0, sdz = 0;
  for (int q = blockIdx.x * NTHREADS + threadIdx.x; q < N4;
       q += gridDim.x * NTHREADS) {
    const int w4 = q % NW4;
    const int t  = q / NW4;
    const int h  = t % Hh;
    const int d  = (t / Hh) % Dd;
    const int idx = 4 * q;                          // element index
    const float4 v = *(const float4*)(flow + idx);
    // axis 4 (W) -> "dz": 3 in-register diffs + 1 seam diff
    sdz += fabsf(v.y - v.x) + fabsf(v.z - v.y) + fabsf(v.w - v.z);
    if (w4 + 1 < NW4) sdz += fabsf(flow[idx + 4] - v.w);
    // axis 3 (H) -> "dx"
    if (h + 1 < Hh) {
      const float4 u = *(const float4*)(flow + idx + Ww);
      sdx += fabsf(u.x - v.x) + fabsf(u.y - v.y) +
             fabsf(u.z - v.z) + fabsf(u.w - v.w);
    }
    // axis 2 (D) -> "dy"
    if (d + 1 < Dd) {
      const float4 u = *(const float4*)(flow + idx + Hh * Ww);
      sdy += fabsf(u.x - v.x) + fabsf(u.y - v.y) +
             fabsf(u.z - v.z) + fabsf(u.w - v.w);
    }
  }
  const float ry = block_reduce_sum(sdy, red);
  const float rx = block_reduce_sum(sdx, red);
  const float rz = block_reduce_sum(sdz, red);
  if (threadIdx.x == 0) {
    dyP[blockIdx.x] = ry;
    dxP[blockIdx.x] = rx;
    dzP[blockIdx.x] = rz;
  }
}

// ---------------------------------------------------------------------------
// Deterministic finalize: fixed-order sum of partials, compute the 3 scalars.
// ---------------------------------------------------------------------------
__global__ __launch_bounds__(NTHREADS)
void finalize_kernel(const float* __restrict__ ccP, int nCc,
                     const float* __restrict__ dyP, const float* __restrict__ dxP,
                     const float* __restrict__ dzP, int nG,
                     float* __restrict__ out) {
  __shared__ float red[NTHREADS];
  const int tid = threadIdx.x;

  float a = 0;
  for (int i = tid; i < nCc; i += NTHREADS) a += ccP[i];
  a = block_reduce_sum(a, red);

  float dy = 0;
  for (int i = tid; i < nG; i += NTHREADS) dy += dyP[i];
  dy = block_reduce_sum(dy, red);

  float dx = 0;
  for (int i = tid; i < nG; i += NTHREADS) dx += dxP[i];
  dx = block_reduce_sum(dx, red);

  float dz = 0;
  for (int i = tid; i < nG; i += NTHREADS) dz += dzP[i];
  dz = block_reduce_sum(dz, red);

  if (tid == 0) {
    const float nvox = (float)((long long)Bn * Dd * Hh * Ww);
    const float L_sim = -(a / nvox);
    const float ny = (float)((long long)Bn * 3 * (Dd - 1) * Hh * Ww);
    const float nx = (float)((long long)Bn * 3 * Dd * (Hh - 1) * Ww);
    const float nz = (float)((long long)Bn * 3 * Dd * Hh * (Ww - 1));
    const float L_reg = dx / nx + dy / ny + dz / nz;
    out[0] = L_sim + L_reg;  // L_total
    out[1] = L_sim;
    out[2] = L_reg;
  }
}

// ---------------------------------------------------------------------------
// Harness entry
// ---------------------------------------------------------------------------
extern "C" void kernel_launch(void* const* d_in, const int* in_sizes, int n_in,
                              void* d_out, int out_size, void* d_ws, size_t ws_size,
                              hipStream_t stream) {
  (void)in_sizes; (void)n_in; (void)out_size; (void)ws_size;
  const float* If   = (const float*)d_in[0];
  const float* Im   = (const float*)d_in[1];
  const float* flow = (const float*)d_in[2];
  float* out = (float*)d_out;
  float* ws  = (float*)d_ws;

  float* ccPart = ws;                           // NCC_BLOCKS   = 600
  float* dyPart = ws + NCC_BLOCKS;              // GRAD_BLOCKS  = 2400
  float* dxPart = dyPart + GRAD_BLOCKS;
  float* dzPart = dxPart + GRAD_BLOCKS;

  const size_t smemBytes = (size_t)SMEM_FLOATS * sizeof(float);  // 236,544 B
  // gfx1250 allows up to 320 KB LDS per workgroup; opt in for >64 KB dyn LDS.
  (void)hipFuncSetAttribute((const void*)lncc_kernel,
                            hipFuncAttributeMaxDynamicSharedMemorySize,
                            (int)smemBytes);

  dim3 grid(Ww / TILE_W, Hh / TILE_H, Bn * ND);  // (5, 6, 20) = 600 blocks
  lncc_kernel<<<grid, NTHREADS, smemBytes, stream>>>(If, Im, ccPart);
  grad_kernel<<<GRAD_BLOCKS, NTHREADS, 0, stream>>>(flow, dyPart, dxPart, dzPart);
  finalize_kernel<<<1, NTHREADS, 0, stream>>>(ccPart, NCC_BLOCKS,
                                              dyPart, dxPart, dzPart, GRAD_BLOCKS,
                                              out);
}